// TokenEmbedding_20985210208416
// MI455X (gfx1250) — compile-verified
//
#include <hip/hip_runtime.h>
#include <hip/hip_bf16.h>

// Problem constants (match reference)
#define B_SZ     32
#define S_LEN    4096
#define C_INCH   21
#define D_MODEL  512
#define KW       8                 // window taps
#define JDIM     (KW * C_INCH)     // 168 fused reduction dim = 42 * 4
#define PROJ_IN  1534              // 73*21 + 1
#define NKER     74

// Tiling
#define DTILE    64                // d per block (4 wmma d-subtiles)
#define WTILE    128               // w per block (8 wmma w-subtiles)
#define XROWS    (WTILE + KW - 1)  // 135 staged x rows
#define WEFF_LD  170               // padded LDS stride (168 -> 170, kills d-row bank conflicts)

typedef __attribute__((ext_vector_type(2))) float v2f;
typedef __attribute__((ext_vector_type(8))) float v8f;

// ---------------------------------------------------------------------------
// Phase 1: fold conv kernels into the projection:
//   W_eff[d, j=k*21+c] = sum_n kernels[n,k]*proj_w[d, n*21+c]
//                        + (c==0)*kernels[73,k]*proj_w[d,1533]
// ---------------------------------------------------------------------------
__global__ __launch_bounds__(256) void build_weff_kernel(
    const float* __restrict__ kernels,  // [74][8]
    const float* __restrict__ proj_w,   // [512][1534]
    float* __restrict__ weff)           // [512][168]
{
    int gid = blockIdx.x * 256 + threadIdx.x;
    if (gid >= D_MODEL * JDIM) return;
    int d = gid / JDIM;
    int j = gid % JDIM;
    int k = j / C_INCH;
    int c = j % C_INCH;
    const float* pw = proj_w + (size_t)d * PROJ_IN;
    float s = 0.0f;
    #pragma unroll 1
    for (int n = 0; n < NKER - 1; ++n)
        s += kernels[n * KW + k] * pw[n * C_INCH + c];
    if (c == 0)
        s += kernels[(NKER - 1) * KW + k] * pw[PROJ_IN - 1];
    weff[gid] = s;
}

// ---------------------------------------------------------------------------
// Phase 2: out[b,d,w] = sum_j xflat[(w-7)*21 + j] * W_eff[d,j] + proj_b[d]
// (TAU=1 makes the im2col column a contiguous 168-float slice of x.)
// Block: 64 d x 128 w. Wave: one 16-d subtile x four 16-w subtiles ->
// each A fragment feeds 4 x V_WMMA_F32_16X16X4_F32.
// ---------------------------------------------------------------------------
__global__ __launch_bounds__(256) void conv_gemm_kernel(
    const float* __restrict__ x,       // [32][4096][21]
    const float* __restrict__ weff,    // [512][168]
    const float* __restrict__ proj_b,  // [512]
    float* __restrict__ out)           // [32][512][4096]
{
    __shared__ float s_weff[DTILE * WEFF_LD];   // 43,520 B
    __shared__ float s_x[XROWS * C_INCH];       // 11,340 B   (total 54,860 B)

    const int blk  = blockIdx.x;
    const int dgrp = blk & 7;            // 8 d-groups of 64
    const int wgrp = (blk >> 3) & 31;    // 32 w-groups of 128
    const int b    = blk >> 8;           // 32 batches

    const int d0 = dgrp * DTILE;
    const int w0 = wgrp * WTILE;
    const int tid = threadIdx.x;

    // Stage W_eff tile (64 x 168) with padded stride. Coalesced global reads.
    for (int i = tid; i < DTILE * JDIM; i += 256) {
        int dl = i / JDIM, j = i - dl * JDIM;
        s_weff[dl * WEFF_LD + j] = weff[(size_t)(d0 + dl) * JDIM + j];
    }

    // Stage raw x rows w0-7 .. w0+127 as one flat contiguous slab (zero-pad
    // the causal front edge). Fully coalesced, no index arithmetic.
    const float* xb = x + (size_t)b * S_LEN * C_INCH;
    const int g0 = (w0 - (KW - 1)) * C_INCH;   // negative only when w0 == 0
    for (int i = tid; i < XROWS * C_INCH; i += 256) {
        int g = g0 + i;
        s_x[i] = (g >= 0) ? xb[g] : 0.0f;
    }
    __syncthreads();

    const int wave  = tid >> 5;      // 0..7
    const int lane  = tid & 31;
    const int half  = lane >> 4;     // K-pair selector (16x4 f32 ISA layout)
    const int lx    = lane & 15;     // M (A) / N (B) index
    const int dsub  = wave & 3;      // which 16-d subtile
    const int whalf = wave >> 2;     // which 64-w half

    v8f acc[4] = {v8f{}, v8f{}, v8f{}, v8f{}};

    // Fold the per-half K offset (2*half) into the base pointers.
    const float* wrow = &s_weff[(dsub * 16 + lx) * WEFF_LD + 2 * half];
    const float* xcol = &s_x[(whalf * 64 + lx) * C_INCH + 2 * half];

    #pragma unroll 3
    for (int jb = 0; jb < JDIM / 4; ++jb) {
        const int ja = jb * 4;
        v2f a;
        a[0] = wrow[ja];
        a[1] = wrow[ja + 1];
        #pragma unroll
        for (int wt = 0; wt < 4; ++wt) {
            v2f bb;
            const float* p = xcol + wt * 16 * C_INCH + ja;
            bb[0] = p[0];
            bb[1] = p[1];
            acc[wt] = __builtin_amdgcn_wmma_f32_16x16x4_f32(
                /*neg_a=*/false, a, /*neg_b=*/false, bb,
                /*c_mod=*/(short)0, acc[wt], /*reuse_a=*/false, /*reuse_b=*/false);
        }
    }

    // Epilogue: bias + store. C/D layout: VGPR v, lane l -> M = v + 8*(l>=16).
    const int dbase = d0 + dsub * 16 + 8 * half;
    float bias[8];
    #pragma unroll
    for (int v = 0; v < 8; ++v) bias[v] = proj_b[dbase + v];

    #pragma unroll
    for (int wt = 0; wt < 4; ++wt) {
        const int wglob = w0 + whalf * 64 + wt * 16 + lx;
        const size_t base = ((size_t)b * D_MODEL + dbase) * S_LEN + wglob;
        #pragma unroll
        for (int v = 0; v < 8; ++v)
            out[base + (size_t)v * S_LEN] = acc[wt][v] + bias[v];
    }
}

// ---------------------------------------------------------------------------
extern "C" void kernel_launch(void* const* d_in, const int* in_sizes, int n_in,
                              void* d_out, int out_size, void* d_ws, size_t ws_size,
                              hipStream_t stream) {
    const float* x       = (const float*)d_in[0];
    const float* kernels = (const float*)d_in[1];
    const float* proj_w  = (const float*)d_in[2];
    const float* proj_b  = (const float*)d_in[3];
    float* out  = (float*)d_out;
    float* weff = (float*)d_ws;          // 512*168*4 = 344,064 bytes scratch

    (void)in_sizes; (void)n_in; (void)out_size; (void)ws_size;

    const int weff_elems = D_MODEL * JDIM;
    build_weff_kernel<<<(weff_elems + 255) / 256, 256, 0, stream>>>(kernels, proj_w, weff);

    const int nblocks = B_SZ * (S_LEN / WTILE) * (D_MODEL / DTILE);  // 8192
    conv_gemm_kernel<<<nblocks, 256, 0, stream>>>(x, weff, proj_b, out);
}